// FFPSSampler_23845658427863
// MI455X (gfx1250) — compile-verified
//
#include <hip/hip_runtime.h>
#include <hip/hip_bf16.h>

typedef __attribute__((ext_vector_type(2))) float v2f;
typedef __attribute__((ext_vector_type(8))) float v8f;

// ---------------------------------------------------------------------------
// Stage 1: build pair-interleaved K-major buffer
//   featP[b][kp][n] = { chan(2kp, n), chan(2kp+1, n) },  kp in [0, KPP=34)
//   chan(k,n): k<3 -> points[b][n][k]; 3<=k<3+C -> features[b][k-3][n]; else 0
// Pair interleave matches the 16x4 f32 WMMA lane layout (each lane consumes
// K-adjacent pairs), so every GEMM operand fetch is one global_load_b64.
// ---------------------------------------------------------------------------
__device__ __forceinline__ float ffps_chan(const float* __restrict__ points,
                                           const float* __restrict__ features,
                                           int b, int k, int n, int N, int C) {
    if (k < 3)          return points[((size_t)b * N + n) * 3 + k];
    if (k < 3 + C)      return features[((size_t)b * C + (k - 3)) * (size_t)N + n];
    return 0.0f;
}

__global__ void ffps_build_featP(const float* __restrict__ points,
                                 const float* __restrict__ features,
                                 float2* __restrict__ featP,
                                 int N, int C, int KPP, long total) {
    long idx = (long)blockIdx.x * blockDim.x + threadIdx.x;
    if (idx >= total) return;
    int n  = (int)(idx % N);
    int kp = (int)((idx / N) % KPP);
    int b  = (int)(idx / ((long)N * KPP));
    float2 v;
    v.x = ffps_chan(points, features, b, 2 * kp,     n, N, C);
    v.y = ffps_chan(points, features, b, 2 * kp + 1, n, N, C);
    featP[idx] = v;
}

// ---------------------------------------------------------------------------
// Stage 2: squared norms sq[b][n] = sum_k feat[k][n]^2 (over pairs)
// ---------------------------------------------------------------------------
__global__ void ffps_sq_norms(const float2* __restrict__ featP,
                              float* __restrict__ sq,
                              int N, int KPP, int total) {
    int idx = blockIdx.x * blockDim.x + threadIdx.x;
    if (idx >= total) return;
    int b = idx / N;
    int n = idx % N;
    const float2* F = featP + (size_t)b * KPP * N + n;
    float s = 0.0f;
    for (int kp = 0; kp < KPP; ++kp) {
        float2 v = F[(size_t)kp * N];
        s += v.x * v.x + v.y * v.y;
    }
    sq[idx] = s;
}

// ---------------------------------------------------------------------------
// Stage 3: Gram matrix G[b] = feat · featᵀ via V_WMMA_F32_16X16X4_F32.
// One wave per 16x16 tile, 8 waves/block. Lane = (m = lane&15, kh = lane>>4):
//   a = {A[m][2kh], A[m][2kh+1]}  = featP[kp = 2*step + kh][rowbase + m]
//   b = {B[2kh][n], B[2kh+1][n]}  = featP[kp = 2*step + kh][colbase + m]
// Base pointers computed once per lane; the fully-unrolled step index gives
// byte offsets step*16*N (64KB/step at N=4096), scaled by the runtime N.
// ---------------------------------------------------------------------------
__global__ void ffps_gram_wmma(const v2f* __restrict__ featP,
                               float* __restrict__ G,
                               int N, int KPP) {
    const int b      = blockIdx.y;
    const int wave   = threadIdx.x >> 5;
    const int lane   = threadIdx.x & 31;
    const int tilesN = N >> 4;
    const int tileId = blockIdx.x * 8 + wave;
    const int rowbase = (tileId / tilesN) << 4;
    const int colbase = (tileId % tilesN) << 4;

    const int m  = lane & 15;
    const int kh = lane >> 4;

    const v2f* Fp = featP + (size_t)b * KPP * N;
    const v2f* pa = Fp + (size_t)kh * N + rowbase + m;
    const v2f* pb = Fp + (size_t)kh * N + colbase + m;

    v8f c = {};
#pragma unroll
    for (int step = 0; step < 17; ++step) {        // KPP == 34 pairs, 2/step
        const v2f a  = pa[(size_t)step * 2 * N];
        const v2f bb = pb[(size_t)step * 2 * N];
        c = __builtin_amdgcn_wmma_f32_16x16x4_f32(
                /*neg_a=*/false, a, /*neg_b=*/false, bb,
                /*c_mod=*/(short)0, c, /*reuse_a=*/false, /*reuse_b=*/false);
    }

    // C layout: VGPR r -> row (r + 8*kh), col (lane&15)
    float* pg = G + (size_t)b * N * N + (size_t)(rowbase + 8 * kh) * N
                  + colbase + m;
#pragma unroll
    for (int r = 0; r < 8; ++r) {
        pg[(size_t)r * N] = c[r];
    }
}

// ---------------------------------------------------------------------------
// Stage 4: sequential FPS. One 1024-thread workgroup per batch.
// min_dist in registers (4 values/thread); per step: one float4 row load of
// G (16KB, L2-resident), register min-update, wave32 shfl_xor argmax,
// 32-wave LDS tree. Tie-break prefers the lower index (jnp.argmax).
// ---------------------------------------------------------------------------
__global__ void __launch_bounds__(1024)
ffps_fps(const float* __restrict__ G,
         const float* __restrict__ sq,
         int* __restrict__ out,
         int N, int npoint) {
    const int b   = blockIdx.x;
    const int tid = threadIdx.x;
    const float* Gb  = G  + (size_t)b * N * N;
    const float* sqb = sq + (size_t)b * N;

    __shared__ float s_val[32];
    __shared__ int   s_idx[32];
    __shared__ int   s_f;

    const int j0 = tid * 4;
    float md0 = 1e10f, md1 = 1e10f, md2 = 1e10f, md3 = 1e10f;
    const float4 sqj = ((const float4*)sqb)[tid];

    int f = 0;
    for (int s = 0; s < npoint; ++s) {
        if (tid == 0) out[(size_t)b * npoint + s] = f;

        const float sqf = sqb[f];
        const float4 g  = ((const float4*)(Gb + (size_t)f * N))[tid];

        md0 = fminf(md0, sqf + sqj.x - 2.0f * g.x);
        md1 = fminf(md1, sqf + sqj.y - 2.0f * g.y);
        md2 = fminf(md2, sqf + sqj.z - 2.0f * g.z);
        md3 = fminf(md3, sqf + sqj.w - 2.0f * g.w);

        float bv = md0; int bi = j0;
        if (md1 > bv) { bv = md1; bi = j0 + 1; }
        if (md2 > bv) { bv = md2; bi = j0 + 2; }
        if (md3 > bv) { bv = md3; bi = j0 + 3; }

#pragma unroll
        for (int off = 16; off >= 1; off >>= 1) {
            float ov = __shfl_xor(bv, off, 32);
            int   oi = __shfl_xor(bi, off, 32);
            if (ov > bv || (ov == bv && oi < bi)) { bv = ov; bi = oi; }
        }

        const int lane = tid & 31, wave = tid >> 5;
        if (lane == 0) { s_val[wave] = bv; s_idx[wave] = bi; }
        __syncthreads();
        if (tid == 0) {
            float best = s_val[0]; int besti = s_idx[0];
            for (int w = 1; w < 32; ++w) {
                if (s_val[w] > best || (s_val[w] == best && s_idx[w] < besti)) {
                    best = s_val[w]; besti = s_idx[w];
                }
            }
            s_f = besti;
        }
        __syncthreads();
        f = s_f;
    }
}

// ---------------------------------------------------------------------------
extern "C" void kernel_launch(void* const* d_in, const int* in_sizes, int n_in,
                              void* d_out, int out_size, void* d_ws, size_t ws_size,
                              hipStream_t stream) {
    const float* points   = (const float*)d_in[0];   // (B, N, 3)
    const float* features = (const float*)d_in[1];   // (B, C, N)

    const int B = 4;
    const int N = in_sizes[0] / (B * 3);             // 4096
    const int C = in_sizes[1] / (B * N);             // 64
    const int KP  = ((3 + C + 3) / 4) * 4;           // 68 (pad K to x4)
    const int KPP = KP / 2;                          // 34 float2 pair-rows
    const int npoint = out_size / B;                 // 1024

    // workspace layout (256B-aligned sections)
    char* ws = (char*)d_ws;
    float2* featP = (float2*)ws;                                     // B*KPP*N
    size_t featP_bytes = ((size_t)B * KPP * N * sizeof(float2) + 255) & ~255ull;
    float* sq = (float*)(ws + featP_bytes);                          // B*N
    size_t sq_bytes = ((size_t)B * N * 4 + 255) & ~255ull;
    float* G = (float*)(ws + featP_bytes + sq_bytes);                // B*N*N
    (void)ws_size;

    int* out = (int*)d_out;

    // 1) pair-interleaved K-major staging buffer
    {
        long total = (long)B * KPP * N;
        int blocks = (int)((total + 255) / 256);
        ffps_build_featP<<<blocks, 256, 0, stream>>>(points, features, featP,
                                                     N, C, KPP, total);
    }
    // 2) squared norms
    {
        int total = B * N;
        ffps_sq_norms<<<(total + 255) / 256, 256, 0, stream>>>(featP, sq,
                                                               N, KPP, total);
    }
    // 3) Gram matrix via WMMA: (N/16)^2 tiles per batch, 8 waves/block
    {
        int tiles = (N / 16) * (N / 16);             // 65536
        dim3 grid(tiles / 8, B);
        ffps_gram_wmma<<<grid, 256, 0, stream>>>((const v2f*)featP, G, N, KPP);
    }
    // 4) sequential FPS, one workgroup per batch
    ffps_fps<<<B, 1024, 0, stream>>>(G, sq, out, N, npoint);
}